// SECGFD_77584289235637
// MI455X (gfx1250) — compile-verified
//
#include <hip/hip_runtime.h>
#include <hip/hip_bf16.h>

typedef __attribute__((ext_vector_type(2))) float v2f;
typedef __attribute__((ext_vector_type(8))) float v8f;

#define NODES 50000
#define EDGES 800000

// ---------------- elementwise / graph kernels ----------------

__global__ void k_zero(float* __restrict__ p, int n) {
  int i = blockIdx.x * 256 + threadIdx.x;
  if (i < n) p[i] = 0.0f;
}

__global__ void k_edge_degree(const int* __restrict__ src, const int* __restrict__ dst,
                              float* __restrict__ deg_in, float* __restrict__ deg_out, int nE) {
  int e = blockIdx.x * 256 + threadIdx.x;
  if (e < nE) {
    atomicAdd(&deg_in[dst[e]], 1.0f);
    atomicAdd(&deg_out[src[e]], 1.0f);
  }
}

__global__ void k_deg_finalize(const float* __restrict__ deg_in, const float* __restrict__ deg_out,
                               float* __restrict__ din, float* __restrict__ dout, int n) {
  int i = blockIdx.x * 256 + threadIdx.x;
  if (i < n) {
    din[i]  = 1.0f / sqrtf(fmaxf(deg_in[i],  1.0f));
    dout[i] = 1.0f / sqrtf(fmaxf(deg_out[i], 1.0f));
  }
}

// y[dst[e]][f] += x[src[e]][f] * scale[src[e]]   (F = 1<<LOGF features)
template <int LOGF>
__global__ void k_spmm_scaled(const float* __restrict__ x, const float* __restrict__ scale,
                              const int* __restrict__ src, const int* __restrict__ dst,
                              float* __restrict__ y, int nE) {
  int t = blockIdx.x * 256 + threadIdx.x;
  if (t >= (nE << LOGF)) return;
  int e = t >> LOGF;
  int f = t & ((1 << LOGF) - 1);
  int s = src[e];
  int d = dst[e];
  atomicAdd(&y[(d << LOGF) + f], x[(s << LOGF) + f] * scale[s]);
}

// out = fin - acc*din[node]   (width 64)
__global__ void k_lap_finalize(const float* __restrict__ fin, const float* __restrict__ acc,
                               const float* __restrict__ din, float* __restrict__ out, int total) {
  int i = blockIdx.x * 256 + threadIdx.x;
  if (i < total) out[i] = fin[i] - acc[i] * din[i >> 6];
}

// g = relu(acc*din[node] + bias[f])   (width 64)
__global__ void k_gcn_fin_relu(const float* __restrict__ acc, const float* __restrict__ din,
                               const float* __restrict__ bias, float* __restrict__ out, int total) {
  int i = blockIdx.x * 256 + threadIdx.x;
  if (i < total) out[i] = fmaxf(acc[i] * din[i >> 6] + bias[i & 63], 0.0f);
}

// emb = acc*din[node] + bias[f]   (width 128)
__global__ void k_gcn_fin(const float* __restrict__ acc, const float* __restrict__ din,
                          const float* __restrict__ bias, float* __restrict__ out, int total) {
  int i = blockIdx.x * 256 + threadIdx.x;
  if (i < total) out[i] = acc[i] * din[i >> 7] + bias[i & 127];
}

// fold the Bernstein theta coefficients into three 64x64 weight blocks
__global__ void k_combine_w3(const float* __restrict__ W3, float* __restrict__ W3c) {
  int i = blockIdx.x * 256 + threadIdx.x;
  if (i < 64 * 64) {
    int k = i >> 6, n = i & 63;
    float a = W3[k * 64 + n];
    float b = W3[(64 + k) * 64 + n];
    float c = W3[(128 + k) * 64 + n];
    float d = W3[(192 + k) * 64 + n];
    float e = W3[(256 + k) * 64 + n];
    W3c[i]             = 3.0f * a;                                    // coeff of h
    W3c[4096 + i]      = -3.0f * a + 3.0f * b + d;                    // coeff of Lh
    W3c[8192 + i]      = 0.75f * a - 1.5f * b + 0.75f * c + e;        // coeff of L2h
  }
}

// out[node,0:2] = t[node,:] @ W4 + b4 ; one wave (32 lanes) per node
__global__ void k_out_head(const float* __restrict__ t, const float* __restrict__ W4,
                           const float* __restrict__ b4, float* __restrict__ out, int n) {
  int wave = (blockIdx.x * 256 + threadIdx.x) >> 5;
  int lane = threadIdx.x & 31;
  if (wave >= n) return;
  const float* tr = t + (size_t)wave * 64;
  float v0 = tr[lane];
  float v1 = tr[lane + 32];
  float s0 = v0 * W4[lane * 2 + 0] + v1 * W4[(lane + 32) * 2 + 0];
  float s1 = v0 * W4[lane * 2 + 1] + v1 * W4[(lane + 32) * 2 + 1];
#pragma unroll
  for (int off = 16; off; off >>= 1) {
    s0 += __shfl_xor(s0, off, 32);
    s1 += __shfl_xor(s1, off, 32);
  }
  if (lane == 0) {
    out[wave * 2 + 0] = s0 + b4[0];
    out[wave * 2 + 1] = s1 + b4[1];
  }
}

// ---------------- WMMA f32 GEMM ----------------
// C[M,N] (+)= A[M,K] @ B[K,N], row-major. N = NT*16, M multiple-of-16 tiles,
// K multiple of 4. EPI: 0 = raw store, 1 = relu(C + bias).
// One wave computes a 16-row strip across all N column tiles using
// V_WMMA_F32_16X16X4_F32; A fragment reused across NT column tiles.
template <int NT, int EPI>
__global__ void k_wmma_gemm(const float* __restrict__ A, const float* __restrict__ B,
                            const float* __restrict__ bias, float* __restrict__ C,
                            int M, int K, int accumulate) {
  const int N = NT * 16;
  int wave = (blockIdx.x * blockDim.x + threadIdx.x) >> 5;  // uniform per wave
  int lane = threadIdx.x & 31;
  if (wave * 16 >= M) return;  // wave-uniform branch, EXEC stays all-1s at WMMA

  int row   = lane & 15;         // A: M index  /  B,C: N index
  int khalf = (lane >> 4) * 2;   // lanes 0-15 -> K 0,1 ; lanes 16-31 -> K 2,3
  int hi    = (lane >> 4) * 8;   // C: lanes 16-31 hold M rows 8..15

  v8f c[NT];
  if (accumulate) {
#pragma unroll
    for (int t = 0; t < NT; ++t)
#pragma unroll
      for (int r = 0; r < 8; ++r)
        c[t][r] = C[(size_t)(wave * 16 + r + hi) * N + t * 16 + row];
  } else {
#pragma unroll
    for (int t = 0; t < NT; ++t)
#pragma unroll
      for (int r = 0; r < 8; ++r)
        c[t][r] = 0.0f;
  }

  const float* Arow = A + (size_t)(wave * 16 + row) * K;
  for (int k = 0; k < K; k += 4) {
    v2f a;
    a.x = Arow[k + khalf];
    a.y = Arow[k + khalf + 1];
#pragma unroll
    for (int t = 0; t < NT; ++t) {
      v2f b;
      b.x = B[(size_t)(k + khalf) * N + t * 16 + row];
      b.y = B[(size_t)(k + khalf + 1) * N + t * 16 + row];
      // 8 args: (neg_a, A, neg_b, B, c_mod, C, reuse_a, reuse_b)
      c[t] = __builtin_amdgcn_wmma_f32_16x16x4_f32(
          false, a, false, b, (short)0, c[t], false, false);
    }
  }

#pragma unroll
  for (int t = 0; t < NT; ++t)
#pragma unroll
    for (int r = 0; r < 8; ++r) {
      int m = wave * 16 + r + hi;
      int n = t * 16 + row;
      float v = c[t][r];
      if (EPI == 1) v = fmaxf(v + bias[n], 0.0f);
      C[(size_t)m * N + n] = v;
    }
}

// ---------------- driver ----------------

static inline int gblk(int n) { return (n + 255) / 256; }

extern "C" void kernel_launch(void* const* d_in, const int* in_sizes, int n_in,
                              void* d_out, int out_size, void* d_ws, size_t ws_size,
                              hipStream_t stream) {
  const float* in_feat = (const float*)d_in[0];
  const int*   src     = (const int*)d_in[1];
  const int*   dst     = (const int*)d_in[2];
  const float* W1 = (const float*)d_in[3],  *b1  = (const float*)d_in[4];
  const float* W2 = (const float*)d_in[5],  *b2  = (const float*)d_in[6];
  const float* W3 = (const float*)d_in[7],  *b3  = (const float*)d_in[8];
  const float* W4 = (const float*)d_in[9],  *b4  = (const float*)d_in[10];
  const float* Wg1 = (const float*)d_in[11], *bg1 = (const float*)d_in[12];
  const float* Wg2 = (const float*)d_in[13], *bg2 = (const float*)d_in[14];

  float* out = (float*)d_out;           // [N,2]
  float* emb = out + NODES * 2;         // [N,128]

  float* ws = (float*)d_ws;
  float* deg_in  = ws; ws += NODES;
  float* deg_out = ws; ws += NODES;
  float* din     = ws; ws += NODES;
  float* dout    = ws; ws += NODES;
  float* h    = ws; ws += (size_t)NODES * 64;
  float* Lh   = ws; ws += (size_t)NODES * 64;
  float* L2h  = ws; ws += (size_t)NODES * 64;
  float* tbuf = ws; ws += (size_t)NODES * 64;   // h1, then t = relu(.@W3+b3)
  float* g    = ws; ws += (size_t)NODES * 64;
  float* tmp  = ws; ws += (size_t)NODES * 128;  // spmm accumulator
  float* p    = ws; ws += (size_t)NODES * 128;  // gcn dense projections
  float* W3c  = ws; ws += 3 * 64 * 64;

  const int gemm_threads = (NODES / 16) * 32;   // 3125 waves
  const int gemm_blocks  = gblk(gemm_threads);

  // degrees -> din/dout
  k_zero<<<gblk(2 * NODES), 256, 0, stream>>>(deg_in, 2 * NODES);
  k_edge_degree<<<gblk(EDGES), 256, 0, stream>>>(src, dst, deg_in, deg_out, EDGES);
  k_deg_finalize<<<gblk(NODES), 256, 0, stream>>>(deg_in, deg_out, din, dout, NODES);

  // MLP front-end: h = relu(relu(in_feat@W1+b1)@W2+b2)
  k_wmma_gemm<4, 1><<<gemm_blocks, 256, 0, stream>>>(in_feat, W1, b1, tbuf, NODES, 128, 0);
  k_wmma_gemm<4, 1><<<gemm_blocks, 256, 0, stream>>>(tbuf, W2, b2, h, NODES, 64, 0);

  // Lh = lap(h), L2h = lap(Lh)
  k_zero<<<gblk(NODES * 64), 256, 0, stream>>>(tmp, NODES * 64);
  k_spmm_scaled<6><<<gblk(EDGES * 64), 256, 0, stream>>>(h, din, src, dst, tmp, EDGES);
  k_lap_finalize<<<gblk(NODES * 64), 256, 0, stream>>>(h, tmp, din, Lh, NODES * 64);

  k_zero<<<gblk(NODES * 64), 256, 0, stream>>>(tmp, NODES * 64);
  k_spmm_scaled<6><<<gblk(EDGES * 64), 256, 0, stream>>>(Lh, din, src, dst, tmp, EDGES);
  k_lap_finalize<<<gblk(NODES * 64), 256, 0, stream>>>(Lh, tmp, din, L2h, NODES * 64);

  // t = relu(h@W3c0 + Lh@W3c1 + L2h@W3c2 + b3)  (theta-folded W3)
  k_combine_w3<<<gblk(64 * 64), 256, 0, stream>>>(W3, W3c);
  k_wmma_gemm<4, 0><<<gemm_blocks, 256, 0, stream>>>(h,   W3c,        nullptr, tbuf, NODES, 64, 0);
  k_wmma_gemm<4, 0><<<gemm_blocks, 256, 0, stream>>>(Lh,  W3c + 4096, nullptr, tbuf, NODES, 64, 1);
  k_wmma_gemm<4, 1><<<gemm_blocks, 256, 0, stream>>>(L2h, W3c + 8192, b3,      tbuf, NODES, 64, 1);

  // out = t@W4 + b4
  k_out_head<<<gblk(NODES * 32), 256, 0, stream>>>(tbuf, W4, b4, out, NODES);

  // GCN branch: gconv(x,W,b) = spmm((x@W)*dout)*din + b
  k_wmma_gemm<4, 0><<<gemm_blocks, 256, 0, stream>>>(in_feat, Wg1, nullptr, p, NODES, 128, 0);
  k_zero<<<gblk(NODES * 64), 256, 0, stream>>>(tmp, NODES * 64);
  k_spmm_scaled<6><<<gblk(EDGES * 64), 256, 0, stream>>>(p, dout, src, dst, tmp, EDGES);
  k_gcn_fin_relu<<<gblk(NODES * 64), 256, 0, stream>>>(tmp, din, bg1, g, NODES * 64);

  k_wmma_gemm<8, 0><<<gemm_blocks, 256, 0, stream>>>(g, Wg2, nullptr, p, NODES, 64, 0);
  k_zero<<<gblk(NODES * 128), 256, 0, stream>>>(tmp, NODES * 128);
  k_spmm_scaled<7><<<gblk(EDGES * 128), 256, 0, stream>>>(p, dout, src, dst, tmp, EDGES);
  k_gcn_fin<<<gblk(NODES * 128), 256, 0, stream>>>(tmp, din, bg2, emb, NODES * 128);
}